// HelixCenterMaskedPriorGenerator_3264175145149
// MI455X (gfx1250) — compile-verified
//
#include <hip/hip_runtime.h>
#include <stdint.h>

// HelixCenterMaskedPriorGenerator for MI455X (gfx1250).
// Store-bandwidth-bound: 402 MB output, ~17.5 us roofline at 23.3 TB/s.
// Per-(b,i) row blocks with 128-wide j tiles; operands staged in LDS
// (legal tile + seq window via CDNA5 async-to-LDS); branchless per-chunk
// decode; 12 fully-coalesced float4 NT stores per thread.

#define KOFF     5
#define NK       11           // 2*K+1
#define MINDIST  3
#define TILE_J   128
#define THREADS  256

typedef float v4f __attribute__((ext_vector_type(4)));
typedef int   v4i __attribute__((ext_vector_type(4)));

typedef __attribute__((address_space(1))) v4i* g_v4i_ptr;  // global
typedef __attribute__((address_space(3))) v4i* l_v4i_ptr;  // LDS
typedef __attribute__((address_space(1))) int* g_i32_ptr;
typedef __attribute__((address_space(3))) int* l_i32_ptr;

__device__ __forceinline__ uint32_t lds_byte_off(const void* p) {
    // generic pointer to LDS: low 32 bits are the LDS byte address
    return (uint32_t)(uintptr_t)p;
}

__device__ __forceinline__ void async_to_lds_b128(uint32_t lds_addr, const float* gsrc) {
#if __has_builtin(__builtin_amdgcn_global_load_async_to_lds_b128)
    __builtin_amdgcn_global_load_async_to_lds_b128(
        (g_v4i_ptr)(uintptr_t)gsrc, (l_v4i_ptr)(uintptr_t)lds_addr, 0, 0);
#else
    asm volatile("global_load_async_to_lds_b128 %0, %1, off"
                 :: "v"(lds_addr), "v"((uint64_t)(uintptr_t)gsrc)
                 : "memory");
#endif
}

__device__ __forceinline__ void async_to_lds_b32(uint32_t lds_addr, const int* gsrc) {
#if __has_builtin(__builtin_amdgcn_global_load_async_to_lds_b32)
    __builtin_amdgcn_global_load_async_to_lds_b32(
        (g_i32_ptr)(uintptr_t)gsrc, (l_i32_ptr)(uintptr_t)lds_addr, 0, 0);
#else
    asm volatile("global_load_async_to_lds_b32 %0, %1, off"
                 :: "v"(lds_addr), "v"((uint64_t)(uintptr_t)gsrc)
                 : "memory");
#endif
}

__device__ __forceinline__ void wait_asynccnt0() {
#if __has_builtin(__builtin_amdgcn_s_wait_asynccnt)
    __builtin_amdgcn_s_wait_asynccnt(0);
#else
    asm volatile("s_wait_asynccnt 0" ::: "memory");
#endif
}

__global__ __launch_bounds__(THREADS)
void helix_prior_kernel(const int* __restrict__ seq,        // (B,L)
                        const float* __restrict__ legal,    // (B,L,L)
                        const float* __restrict__ emb,      // (5,4)
                        const unsigned char* __restrict__ can, // (5,5) bool
                        float* __restrict__ out,            // (B,L,L,96)
                        int B, int L)
{
    __shared__ float    s_legal[TILE_J];               // legal[b,i,j0:j0+128]
    __shared__ v4f      s_emb[5];                      // emb rows (row 4 = zeros)
    __shared__ int      s_seqj[TILE_J + 2 * KOFF + 1]; // seq window around j tile
    __shared__ uint64_t s_bpack;                       // 11 bases of row i, 3 bits each
    __shared__ uint32_t s_can;                         // packed 25-bit canonical table

    const int tid = threadIdx.x;
    const int j0  = blockIdx.x * TILE_J;
    const int i   = blockIdx.y;
    const int b   = blockIdx.z;

    const int* seq_b = seq + (size_t)b * L;
    const bool full_tile = (j0 + TILE_J) <= L;
    const int  seq_win   = TILE_J + 2 * KOFF + 1;                 // 139
    const bool interior  = (j0 >= KOFF) && (j0 + TILE_J + KOFF + 1 <= L);

    // ---- stage legal_mask tile: async global->LDS (512 B, one wave, b128) ----
    if (full_tile) {
        if (tid < 32) {
            const float* gsrc = legal + ((size_t)(b * L + i) * L + j0) + (size_t)tid * 4;
            async_to_lds_b128(lds_byte_off(&s_legal[tid * 4]), gsrc);
        }
    } else {
        for (int t = tid; t < TILE_J; t += THREADS) {
            int j = j0 + t;
            s_legal[t] = (j < L) ? legal[(size_t)(b * L + i) * L + j] : 0.0f;
        }
    }

    // ---- stage seq window (positions j0-5 .. j0+TILE_J+5) ----
    if (interior) {
        if (tid < seq_win)
            async_to_lds_b32(lds_byte_off(&s_seqj[tid]), seq_b + (j0 - KOFF + tid));
    } else {
        for (int t = tid; t < seq_win; t += THREADS) {
            int p = j0 - KOFF + t;
            s_seqj[t] = (p >= 0 && p < L) ? seq_b[p] : 4;
        }
    }

    // ---- emb rows as float4 ----
    if (tid < 5) s_emb[tid] = ((const v4f*)emb)[tid];

    // ---- pack row-i bases (3b each) and canonical table (1b each) ----
    if (tid == 0) {
        uint64_t bp = 0;
        #pragma unroll
        for (int k = 0; k < NK; ++k) {
            int p = i - KOFF + k;
            uint64_t v = (p >= 0 && p < L) ? (uint64_t)seq_b[p] : 4ull;
            bp |= v << (3 * k);
        }
        s_bpack = bp;
        uint32_t m = 0;
        #pragma unroll
        for (int t = 0; t < 25; ++t) m |= (can[t] ? 1u : 0u) << t;
        s_can = m;
    }

    wait_asynccnt0();        // no-op for waves with ASYNCcnt==0
    __syncthreads();

    const uint32_t canm  = s_can;
    const uint64_t bpack = s_bpack;

    // Output slab (b,i,j0:j0+TILE_J,0:96) is contiguous: 24 float4 per j.
    v4f* __restrict__ out4 = (v4f*)out;
    const size_t outbase = ((size_t)(b * L + i) * L + j0) * 24;

    #pragma unroll
    for (int it = 0; it < (TILE_J * 24) / THREADS; ++it) {
        const int off = it * THREADS + tid;     // float4 index in tile
        const int jl  = off / 24;               // j - j0   (const-div -> mulhi)
        const int c4  = off - jl * 24;          // chunk within 96 channels (0..23)
        const int j   = j0 + jl;
        if (j >= L) continue;                   // partial-tile guard (L%TILE_J only)

        // Branchless decode: chunks 0/1 (base_info) are the kk=KOFF diagonal
        // with v=1; chunks 2..23 are (k,half) feature blocks scaled by the
        // masked legal value.
        const int  half = (c4 - 2) & 1;                        // 0->row base, 1->col base
        const int  kk   = (c4 < 2) ? KOFF : ((c4 - 2) >> 1);   // diagonal index 0..10
        const int  bi   = (int)((bpack >> (3 * kk)) & 7u);     // seq[i+kk-5] (OOB->4)
        const int  bj   = s_seqj[jl + 2 * KOFF - kk];          // seq[j-(kk-5)] (OOB->4)
        const int  md   = (j - i) > (MINDIST + 2 * (kk - KOFF));
        const int  mc   = (canm >> (bi * 5 + bj)) & 1;
        const float mlv = (md & mc) ? s_legal[jl] : 0.0f;
        const float v   = (c4 < 2) ? 1.0f : mlv;
        const v4f   e   = s_emb[half ? bj : bi];

        // streaming NT store: output (402 MB) exceeds L2, don't pollute it
        __builtin_nontemporal_store(e * v, &out4[outbase + off]);
    }
}

extern "C" void kernel_launch(void* const* d_in, const int* in_sizes, int n_in,
                              void* d_out, int out_size, void* d_ws, size_t ws_size,
                              hipStream_t stream) {
    const int*           seq   = (const int*)d_in[0];
    const float*         legal = (const float*)d_in[1];
    const float*         emb   = (const float*)d_in[2];
    const unsigned char* can   = (const unsigned char*)d_in[3];
    float*               out   = (float*)d_out;

    const int BL  = in_sizes[0];           // B*L
    const int BLL = in_sizes[1];           // B*L*L
    const int L   = BLL / BL;
    const int B   = BL / L;

    dim3 grid((L + TILE_J - 1) / TILE_J, L, B);
    helix_prior_kernel<<<grid, THREADS, 0, stream>>>(seq, legal, emb, can, out, B, L);
}